// myModel1_7713761264311
// MI455X (gfx1250) — compile-verified
//
#include <hip/hip_runtime.h>

// ---------------------------------------------------------------------------
// MI455X (gfx1250) implementation. wave32. WMMA f32 16x16x4 for all GEMMs.
// Dims (hardcoded per reference): N=100000 M=50000 D=64 B=3 L=2
// ---------------------------------------------------------------------------

typedef float v2f __attribute__((ext_vector_type(2)));
typedef float v8f __attribute__((ext_vector_type(8)));

#define MODE_MEAN3_SIG 0
#define MODE_SIG 1
#define MODE_ACC 2

__device__ __forceinline__ v8f wmma4(v2f a, v2f b, v8f c) {
  // D = A(16x4,f32) * B(4x16,f32) + C(16x16,f32)
  return __builtin_amdgcn_wmma_f32_16x16x4_f32(
      /*neg_a=*/false, a, /*neg_b=*/false, b,
      /*c_mod=*/(short)0, c, /*reuse_a=*/false, /*reuse_b=*/false);
}

// Stage W[K][64] into LDS transposed: ldsWT[n*K + k] = W[k*64 + n]
template <int K>
__device__ __forceinline__ void load_wt(const float* __restrict__ W, float* lds) {
  for (int idx = threadIdx.x; idx < K * 64; idx += blockDim.x) {
    int k = idx >> 6, n = idx & 63;
    lds[n * K + k] = W[idx];
  }
  __syncthreads();
}

// One wave computes a 16x64 output tile. afr = K/4 fp32 A-fragments.
// A-frag layout (ISA 7.12.2): lanes0-15 hold (K=4kb,4kb+1), lanes16-31 (4kb+2,4kb+3).
// B-frag: b.x = W[k0][n], b.y = W[k0+1][n]  (contiguous in the LDS transpose).
template <int K>
__device__ __forceinline__ void gemm_tile(const v2f* afr, const float* ldsWT,
                                          int lane, v8f acc[4]) {
  const int hi = (lane >> 4) & 1;
  const int n15 = lane & 15;
#pragma unroll
  for (int j = 0; j < 4; ++j) {
    v8f c = acc[j];
    const float* wt = ldsWT + (size_t)(j * 16 + n15) * K + 2 * hi;
#pragma unroll
    for (int kb = 0; kb < K / 4; ++kb) {
      v2f b = *(const v2f*)(wt + kb * 4);
      c = wmma4(afr[kb], b, c);
    }
    acc[j] = c;
  }
}

// ---------------------------------------------------------------------------
// SpMM: out[r] += v * x[c], 16 threads/edge, float4 per thread, f32 atomics
// (accumulators are L2-resident; edge arrays stream from HBM).
// ---------------------------------------------------------------------------
__global__ void spmm_kernel(const int* __restrict__ rows, const int* __restrict__ cols,
                            const float* __restrict__ vals, const float* __restrict__ x,
                            float* __restrict__ out, int nE) {
  long long t = (long long)blockIdx.x * blockDim.x + threadIdx.x;
  int e = (int)(t >> 4);
  if (e >= nE) return;
  __builtin_prefetch(cols + e + 4096, 0, 1);
  int sub = ((int)t & 15) * 4;
  int r = rows[e], c = cols[e];
  float v = vals[e];
  const float4 xv = *(const float4*)(x + (size_t)c * 64 + sub);
  float* o = out + (size_t)r * 64 + sub;
  atomicAdd(o + 0, v * xv.x);
  atomicAdd(o + 1, v * xv.y);
  atomicAdd(o + 2, v * xv.z);
  atomicAdd(o + 3, v * xv.w);
}

// ---------------------------------------------------------------------------
// rs[row] = sum_d leaky_relu_0.2( ((it ⊙ nn) @ W)[row, d] )
// (x_in / x_out are only ever consumed via this row-sum)
// ---------------------------------------------------------------------------
__global__ void score_kernel(const float* __restrict__ it, const float* __restrict__ nn,
                             const float* __restrict__ W, float* __restrict__ rs) {
  __shared__ float wt[64 * 64];
  load_wt<64>(W, wt);
  const int lane = threadIdx.x & 31;
  const int hi = lane >> 4, m15 = lane & 15, koff = hi * 2;
  const int row0 = blockIdx.x * 16;
  const float* itr = it + (size_t)(row0 + m15) * 64 + koff;
  const float* nnr = nn + (size_t)(row0 + m15) * 64 + koff;
  v2f afr[16];
#pragma unroll
  for (int kb = 0; kb < 16; ++kb) {
    v2f a = *(const v2f*)(itr + kb * 4);
    v2f g = *(const v2f*)(nnr + kb * 4);
    v2f f; f.x = a.x * g.x; f.y = a.y * g.y;
    afr[kb] = f;
  }
  v8f acc[4] = {};
  gemm_tile<64>(afr, wt, lane, acc);
  float p[8];
#pragma unroll
  for (int v = 0; v < 8; ++v) {
    float s = 0.f;
#pragma unroll
    for (int j = 0; j < 4; ++j) {
      float x = acc[j][v];
      s += (x > 0.f) ? x : 0.2f * x;
    }
    p[v] = s;
  }
  // Reduce across the 16 lanes of each half (row m = v + 8*hi lives in one half).
#pragma unroll
  for (int v = 0; v < 8; ++v) {
    float s = p[v];
    s += __shfl_xor(s, 1, 32);
    s += __shfl_xor(s, 2, 32);
    s += __shfl_xor(s, 4, 32);
    s += __shfl_xor(s, 8, 32);
    p[v] = s;
  }
  if (m15 == 0) {
#pragma unroll
    for (int v = 0; v < 8; ++v) rs[row0 + hi * 8 + v] = p[v];
  }
}

// ---------------------------------------------------------------------------
// u_stk[b] = concat(u_ui, u_gnn) @ uu_w ; u_gnn fused: nb_u*cw0 + u*cw1 + cb
// ---------------------------------------------------------------------------
__global__ void user_stk_kernel(const float* __restrict__ u_ui, const float* __restrict__ nb_u,
                                const float* __restrict__ u, const float* __restrict__ cw,
                                const float* __restrict__ cb, const float* __restrict__ W,
                                float* __restrict__ out) {
  __shared__ float wt[128 * 64];
  load_wt<128>(W, wt);
  const int lane = threadIdx.x & 31;
  const int hi = lane >> 4, m15 = lane & 15, koff = hi * 2, n15 = m15;
  const int row0 = blockIdx.x * 16;
  const size_t rb = (size_t)(row0 + m15) * 64 + koff;
  const float cw0 = cw[0], cw1 = cw[1], cbv = cb[0];
  v2f afr[32];
#pragma unroll
  for (int kb = 0; kb < 16; ++kb) afr[kb] = *(const v2f*)(u_ui + rb + kb * 4);
#pragma unroll
  for (int kb = 0; kb < 16; ++kb) {
    v2f nv = *(const v2f*)(nb_u + rb + kb * 4);
    v2f uv = *(const v2f*)(u + rb + kb * 4);
    v2f f; f.x = nv.x * cw0 + uv.x * cw1 + cbv; f.y = nv.y * cw0 + uv.y * cw1 + cbv;
    afr[16 + kb] = f;
  }
  v8f acc[4] = {};
  gemm_tile<128>(afr, wt, lane, acc);
#pragma unroll
  for (int j = 0; j < 4; ++j)
#pragma unroll
    for (int v = 0; v < 8; ++v)
      out[(size_t)(row0 + hi * 8 + v) * 64 + j * 16 + n15] = acc[j][v];
}

// ---------------------------------------------------------------------------
// i_stk[b] = concat(i_gnn, i_ui) @ ii_w
// i_gnn fused: it*cw0 + (in_n*s0 + out_n*s1)*cw1 + cb, s = softmax(rs*0.125)
// ---------------------------------------------------------------------------
__global__ void item_stk_kernel(const float* __restrict__ it, const float* __restrict__ in_n,
                                const float* __restrict__ out_n, const float* __restrict__ i_ui,
                                const float* __restrict__ rs_in, const float* __restrict__ rs_out,
                                const float* __restrict__ cw, const float* __restrict__ cb,
                                const float* __restrict__ W, float* __restrict__ out) {
  __shared__ float wt[128 * 64];
  load_wt<128>(W, wt);
  const int lane = threadIdx.x & 31;
  const int hi = lane >> 4, m15 = lane & 15, koff = hi * 2, n15 = m15;
  const int row0 = blockIdx.x * 16;
  const int m = row0 + m15;
  const size_t rb = (size_t)m * 64 + koff;
  float e0 = rs_in[m] * 0.125f, e1 = rs_out[m] * 0.125f;
  float mx = fmaxf(e0, e1);
  float x0 = __expf(e0 - mx), x1 = __expf(e1 - mx);
  float rcp = 1.f / (x0 + x1);
  float s0 = x0 * rcp, s1 = x1 * rcp;
  const float cw0 = cw[0], cw1 = cw[1], cbv = cb[0];
  v2f afr[32];
#pragma unroll
  for (int kb = 0; kb < 16; ++kb) {
    v2f iv = *(const v2f*)(it + rb + kb * 4);
    v2f an = *(const v2f*)(in_n + rb + kb * 4);
    v2f ao = *(const v2f*)(out_n + rb + kb * 4);
    v2f f;
    f.x = iv.x * cw0 + (an.x * s0 + ao.x * s1) * cw1 + cbv;
    f.y = iv.y * cw0 + (an.y * s0 + ao.y * s1) * cw1 + cbv;
    afr[kb] = f;
  }
#pragma unroll
  for (int kb = 0; kb < 16; ++kb) afr[16 + kb] = *(const v2f*)(i_ui + rb + kb * 4);
  v8f acc[4] = {};
  gemm_tile<128>(afr, wt, lane, acc);
#pragma unroll
  for (int j = 0; j < 4; ++j)
#pragma unroll
    for (int v = 0; v < 8; ++v)
      out[(size_t)(row0 + hi * 8 + v) * 64 + j * 16 + n15] = acc[j][v];
}

// ---------------------------------------------------------------------------
// Generic K=64 GEMM with epilogue:
//  MEAN3_SIG: out = sigmoid(((A0+A1+A2)/3) @ W)
//  SIG:       out = sigmoid(A0 @ W)          (in-place safe: out may == A0)
//  ACC:       out += A0 @ W
// ---------------------------------------------------------------------------
__global__ void gemm64_epi_kernel(const float* __restrict__ A0, const float* __restrict__ A1,
                                  const float* __restrict__ A2, const float* __restrict__ W,
                                  float* __restrict__ out, int mode) {
  __shared__ float wt[64 * 64];
  load_wt<64>(W, wt);
  const int lane = threadIdx.x & 31;
  const int hi = lane >> 4, m15 = lane & 15, koff = hi * 2, n15 = m15;
  const int row0 = blockIdx.x * 16;
  const size_t rb = (size_t)(row0 + m15) * 64 + koff;
  v2f afr[16];
  if (mode == MODE_MEAN3_SIG) {
#pragma unroll
    for (int kb = 0; kb < 16; ++kb) {
      v2f a = *(const v2f*)(A0 + rb + kb * 4);
      v2f b = *(const v2f*)(A1 + rb + kb * 4);
      v2f c = *(const v2f*)(A2 + rb + kb * 4);
      v2f f; f.x = (a.x + b.x + c.x) * (1.f / 3.f); f.y = (a.y + b.y + c.y) * (1.f / 3.f);
      afr[kb] = f;
    }
  } else {
#pragma unroll
    for (int kb = 0; kb < 16; ++kb) afr[kb] = *(const v2f*)(A0 + rb + kb * 4);
  }
  v8f acc[4] = {};
  gemm_tile<64>(afr, wt, lane, acc);
#pragma unroll
  for (int j = 0; j < 4; ++j)
#pragma unroll
    for (int v = 0; v < 8; ++v) {
      size_t idx = (size_t)(row0 + hi * 8 + v) * 64 + j * 16 + n15;
      float x = acc[j][v];
      if (mode == MODE_ACC) out[idx] += x;
      else out[idx] = 1.f / (1.f + __expf(-x));
    }
}

// ---------------------------------------------------------------------------
// Host orchestration
// ---------------------------------------------------------------------------
extern "C" void kernel_launch(void* const* d_in, const int* in_sizes, int n_in,
                              void* d_out, int out_size, void* d_ws, size_t ws_size,
                              hipStream_t stream) {
  (void)in_sizes; (void)n_in; (void)ws_size;
  const int N = 100000, M = 50000, Bc = 3, Lc = 2;
  const int E_UI = 1600000, E_II = 800000, E_UU = 1600000;

  const float* user_emb = (const float*)d_in[0];
  const float* item_emb = (const float*)d_in[1];
  const int* ui_rows = (const int*)d_in[2];
  const int* ui_cols = (const int*)d_in[3];
  const float* ui_vals = (const float*)d_in[4];
  const int* uit_rows = (const int*)d_in[5];
  const int* uit_cols = (const int*)d_in[6];
  const float* uit_vals = (const float*)d_in[7];
  const int* ii_in_rows = (const int*)d_in[8];
  const int* ii_in_cols = (const int*)d_in[9];
  const float* ii_in_vals = (const float*)d_in[10];
  const int* ii_out_rows = (const int*)d_in[11];
  const int* ii_out_cols = (const int*)d_in[12];
  const float* ii_out_vals = (const float*)d_in[13];
  const int* uu_rows = (const int*)d_in[14];
  const int* uu_cols = (const int*)d_in[15];
  const float* uu_vals = (const float*)d_in[16];
  const float* i_w = (const float*)d_in[17];
  const float* u_w = (const float*)d_in[18];
  const float* ii_w = (const float*)d_in[19];
  const float* uu_w = (const float*)d_in[20];
  const float* W1 = (const float*)d_in[21];
  const float* W2 = (const float*)d_in[22];
  const float* conv_w = (const float*)d_in[23];
  const float* conv_b = (const float*)d_in[24];
  const float* conv_user_w = (const float*)d_in[25];
  const float* conv_user_b = (const float*)d_in[26];
  const float* u_conc_w = (const float*)d_in[27];
  const float* i_conc_w = (const float*)d_in[28];

  // Workspace carve-up (floats)
  float* ws = (float*)d_ws;
  size_t off = 0;
  auto alloc = [&](size_t n) { float* p = ws + off; off += n; return p; };
  float* uA = alloc((size_t)N * 64);
  float* uB = alloc((size_t)N * 64);
  float* iA = alloc((size_t)M * 64);
  float* iB = alloc((size_t)M * 64);
  float* u_stk = alloc((size_t)Bc * N * 64);
  float* i_stk = alloc((size_t)Bc * M * 64);
  float* nb_u = alloc((size_t)N * 64);
  float* in_n = alloc((size_t)M * 64);
  float* out_n = alloc((size_t)M * 64);
  float* u_ui = alloc((size_t)N * 64);
  float* i_ui = alloc((size_t)M * 64);
  float* rs_in = alloc((size_t)M);
  float* rs_out = alloc((size_t)M);

  // Output carve-up: uo[N,64] | io[M,64] | uos[B,N,64] | ios[B,M,64]
  float* uo = (float*)d_out;
  float* io = uo + (size_t)N * 64;
  float* uos = io + (size_t)M * 64;
  float* ios = uos + (size_t)Bc * N * 64;

  hipMemsetAsync(d_out, 0, (size_t)out_size * sizeof(float), stream);
  hipMemcpyAsync(uA, user_emb, (size_t)N * 64 * sizeof(float), hipMemcpyDeviceToDevice, stream);
  hipMemcpyAsync(iA, item_emb, (size_t)M * 64 * sizeof(float), hipMemcpyDeviceToDevice, stream);

  float* ucur = uA; float* unext = uB;
  float* icur = iA; float* inext = iB;
  const int NT_N = N / 16, NT_M = M / 16;  // 6250 / 3125 — exact

  auto spmm = [&](const int* r, const int* c, const float* v, const float* x, float* o, int nE) {
    long long tot = (long long)nE * 16;
    int blocks = (int)((tot + 255) / 256);
    spmm_kernel<<<blocks, 256, 0, stream>>>(r, c, v, x, o, nE);
  };

  for (int l = 0; l < Lc; ++l) {
    for (int b = 0; b < Bc; ++b) {
      hipMemsetAsync(nb_u, 0, (size_t)N * 64 * sizeof(float), stream);
      hipMemsetAsync(in_n, 0, (size_t)M * 64 * sizeof(float), stream);
      hipMemsetAsync(out_n, 0, (size_t)M * 64 * sizeof(float), stream);
      hipMemsetAsync(u_ui, 0, (size_t)N * 64 * sizeof(float), stream);
      hipMemsetAsync(i_ui, 0, (size_t)M * 64 * sizeof(float), stream);

      spmm(uu_rows + (size_t)b * E_UU, uu_cols + (size_t)b * E_UU, uu_vals + (size_t)b * E_UU,
           ucur, nb_u, E_UU);
      spmm(ii_in_rows + (size_t)b * E_II, ii_in_cols + (size_t)b * E_II, ii_in_vals + (size_t)b * E_II,
           icur, in_n, E_II);
      spmm(ii_out_rows + (size_t)b * E_II, ii_out_cols + (size_t)b * E_II, ii_out_vals + (size_t)b * E_II,
           icur, out_n, E_II);
      spmm(ui_rows + (size_t)b * E_UI, ui_cols + (size_t)b * E_UI, ui_vals + (size_t)b * E_UI,
           icur, u_ui, E_UI);
      spmm(uit_rows + (size_t)b * E_UI, uit_cols + (size_t)b * E_UI, uit_vals + (size_t)b * E_UI,
           ucur, i_ui, E_UI);

      score_kernel<<<NT_M, 32, 0, stream>>>(icur, in_n, W1 + (size_t)l * 64 * 64, rs_in);
      score_kernel<<<NT_M, 32, 0, stream>>>(icur, out_n, W2 + (size_t)l * 64 * 64, rs_out);

      user_stk_kernel<<<NT_N, 32, 0, stream>>>(u_ui, nb_u, ucur, conv_user_w + l * 2,
                                               conv_user_b + l, uu_w + (size_t)l * 128 * 64,
                                               u_stk + (size_t)b * N * 64);
      item_stk_kernel<<<NT_M, 32, 0, stream>>>(icur, in_n, out_n, i_ui, rs_in, rs_out,
                                               conv_w + l * 2, conv_b + l,
                                               ii_w + (size_t)l * 128 * 64,
                                               i_stk + (size_t)b * M * 64);
    }

    const float* uwl = u_w + (size_t)l * 64 * 64;
    const float* iwl = i_w + (size_t)l * 64 * 64;
    const float* ucb = u_conc_w + (size_t)l * 64 * 64;  // rows [l*64, l*64+64) of [128,64]
    const float* icb = i_conc_w + (size_t)l * 64 * 64;

    // u_next = sigmoid(mean_b(u_stk) @ u_w[l]) ; likewise items (reads ORIGINAL stk)
    gemm64_epi_kernel<<<NT_N, 32, 0, stream>>>(u_stk, u_stk + (size_t)N * 64,
                                               u_stk + (size_t)2 * N * 64, uwl, unext, MODE_MEAN3_SIG);
    gemm64_epi_kernel<<<NT_M, 32, 0, stream>>>(i_stk, i_stk + (size_t)M * 64,
                                               i_stk + (size_t)2 * M * 64, iwl, inext, MODE_MEAN3_SIG);
    // uo += u_next @ u_conc_w[l-block] (concat-over-L GEMM decomposed into accumulates)
    gemm64_epi_kernel<<<NT_N, 32, 0, stream>>>(unext, nullptr, nullptr, ucb, uo, MODE_ACC);
    gemm64_epi_kernel<<<NT_M, 32, 0, stream>>>(inext, nullptr, nullptr, icb, io, MODE_ACC);

    for (int b = 0; b < Bc; ++b) {
      float* usb = u_stk + (size_t)b * N * 64;
      float* isb = i_stk + (size_t)b * M * 64;
      // stk[b] <- sigmoid(stk[b] @ w[l])  (in-place), then uos[b]/ios[b] += stk[b] @ conc-block
      gemm64_epi_kernel<<<NT_N, 32, 0, stream>>>(usb, nullptr, nullptr, uwl, usb, MODE_SIG);
      gemm64_epi_kernel<<<NT_M, 32, 0, stream>>>(isb, nullptr, nullptr, iwl, isb, MODE_SIG);
      gemm64_epi_kernel<<<NT_N, 32, 0, stream>>>(usb, nullptr, nullptr, ucb,
                                                 uos + (size_t)b * N * 64, MODE_ACC);
      gemm64_epi_kernel<<<NT_M, 32, 0, stream>>>(isb, nullptr, nullptr, icb,
                                                 ios + (size_t)b * M * 64, MODE_ACC);
    }

    float* t = ucur; ucur = unext; unext = t;
    t = icur; icur = inext; inext = t;
  }
}